// LinODECell_19988777796139
// MI455X (gfx1250) — compile-verified
//
#include <hip/hip_runtime.h>

typedef __attribute__((ext_vector_type(2))) float v2f;
typedef __attribute__((ext_vector_type(4))) float v4f;
typedef __attribute__((ext_vector_type(8))) float v8f;

#define NTERMS   20   // Taylor terms: ||A*dt|| <~ 2 -> remainder ~1e-13, below fp32 eps
#define COLS     16   // batch columns per workgroup (one WMMA N-tile)
#define DDIM     64   // latent dimension
#define LDSTRIDE 68   // padded column stride (floats) -> conflict-free ds_load_b64

__global__ __launch_bounds__(128) void linode_expm_action_kernel(
    const float* __restrict__ dt,   // [B]
    const float* __restrict__ x0,   // [B, 64] row-major
    const float* __restrict__ A,    // [64, 64] row-major
    float* __restrict__ out)        // [B, 64] row-major
{
    __shared__ float buf0[COLS * LDSTRIDE];
    __shared__ float buf1[COLS * LDSTRIDE];

    const int tid  = threadIdx.x;
    const int lane = tid & 31;
    const int wave = tid >> 5;            // row block: rows [16*wave, 16*wave+16)
    const int c    = lane & 15;           // column (batch element) within tile
    const int hi   = (lane >> 4) & 1;     // lane group 0: lanes 0-15, 1: lanes 16-31
    const int colbase = blockIdx.x * COLS;

    // ---- cooperative coalesced load: x0 tile (contiguous 4KB) -> buf0 column-major ----
    {
        const v4f* gx = (const v4f*)(x0 + (size_t)colbase * DDIM);
        #pragma unroll
        for (int r = 0; r < 2; ++r) {
            int i4 = tid + r * 128;       // 256 float4 chunks per tile
            int cc = i4 >> 4;             // column
            int j4 = i4 & 15;             // float4 index within column
            *(v4f*)&buf0[cc * LDSTRIDE + j4 * 4] = gx[i4];
        }
    }

    // ---- per-lane dt (column scalar; duplicated across the two lane groups) ----
    const float dt_l = dt[colbase + c];

    // ---- A fragments, resident in VGPRs (16 rows x 64 cols per wave = 32 VGPRs) ----
    // ISA 16x4 fp32 A layout: lanes 0-15 -> M=lane, K=4kt+{0,1};
    //                         lanes 16-31 -> M=lane-16, K=4kt+{2,3}
    v2f a[16];
    {
        const int r  = wave * 16 + c;
        const int co = hi * 2;
        #pragma unroll
        for (int kt = 0; kt < 16; ++kt)
            a[kt] = *(const v2f*)(A + r * DDIM + 4 * kt + co);
    }

    __syncthreads();

    // ---- sum = t0 (C/D layout: VGPR v -> row wave*16 + v + 8*hi, col c) ----
    v8f sum;
    {
        const float* p = &buf0[c * LDSTRIDE + wave * 16 + hi * 8];
        v4f lo = *(const v4f*)p;
        v4f h4 = *(const v4f*)(p + 4);
        sum[0] = lo[0]; sum[1] = lo[1]; sum[2] = lo[2]; sum[3] = lo[3];
        sum[4] = h4[0]; sum[5] = h4[1]; sum[6] = h4[2]; sum[7] = h4[3];
    }

    // ---- Taylor iterations: t_{k} = (dt/k) * A * t_{k-1}, ping-pong LDS buffers ----
    #pragma unroll
    for (int it = 1; it <= NTERMS; ++it) {
        const float* rb = ((it & 1) != 0) ? buf0 : buf1;  // read t_{it-1}
        float*       wb = ((it & 1) != 0) ? buf1 : buf0;  // write t_{it}

        v8f acc = {};
        #pragma unroll
        for (int kt = 0; kt < 16; ++kt) {
            // B-fragment 4x16: lanes 0-15 -> K=4kt+{0,1}, lanes 16-31 -> K=4kt+{2,3}
            v2f b = *(const v2f*)&rb[c * LDSTRIDE + 4 * kt + hi * 2];
            acc = __builtin_amdgcn_wmma_f32_16x16x4_f32(
                false, a[kt], false, b, (short)0, acc, false, false);
        }

        const float s = dt_l * (1.0f / (float)it);   // constant 1/it folds (full unroll)
        #pragma unroll
        for (int v = 0; v < 8; ++v) { acc[v] *= s; sum[v] += acc[v]; }

        // store t_it (C/D layout -> column-major LDS), two ds_store_b128 per lane
        {
            float* p = &wb[c * LDSTRIDE + wave * 16 + hi * 8];
            v4f lo = { acc[0], acc[1], acc[2], acc[3] };
            v4f h4 = { acc[4], acc[5], acc[6], acc[7] };
            *(v4f*)p       = lo;
            *(v4f*)(p + 4) = h4;
        }
        __syncthreads();
    }

    // ---- deposit sum into buf0, then cooperative coalesced store to out ----
    {
        float* p = &buf0[c * LDSTRIDE + wave * 16 + hi * 8];
        v4f lo = { sum[0], sum[1], sum[2], sum[3] };
        v4f h4 = { sum[4], sum[5], sum[6], sum[7] };
        *(v4f*)p       = lo;
        *(v4f*)(p + 4) = h4;
    }
    __syncthreads();

    {
        v4f* gout = (v4f*)(out + (size_t)colbase * DDIM);
        #pragma unroll
        for (int r = 0; r < 2; ++r) {
            int i4 = tid + r * 128;
            int cc = i4 >> 4;
            int j4 = i4 & 15;
            gout[i4] = *(const v4f*)&buf0[cc * LDSTRIDE + j4 * 4];
        }
    }
}

extern "C" void kernel_launch(void* const* d_in, const int* in_sizes, int n_in,
                              void* d_out, int out_size, void* d_ws, size_t ws_size,
                              hipStream_t stream) {
    (void)n_in; (void)out_size; (void)d_ws; (void)ws_size;
    const float* dt     = (const float*)d_in[0];   // [B]
    const float* x0     = (const float*)d_in[1];   // [B, 64]
    const float* kernel = (const float*)d_in[2];   // [64, 64]
    float* out = (float*)d_out;                    // [B, 64]

    const int B = in_sizes[0];
    const int grid = B / COLS;                     // 16384 / 16 = 1024 workgroups
    linode_expm_action_kernel<<<grid, 128, 0, stream>>>(dt, x0, kernel, out);
}